// LayoutDiscriminator_13503377178745
// MI455X (gfx1250) — compile-verified
//
#include <hip/hip_runtime.h>
#include <hip/hip_bf16.h>
#include <math.h>
#include <stdint.h>

// ---- problem constants ----
#define B_  32
#define N_  64
#define CC_ 16
#define HH_ 63
#define WW_ 63

// A-tile K-chunk staged in LDS (16 rows x KC_ cols x 4B = 36 KB)
#define KC_ 576

// Use gfx1250 async global->LDS copy for the weight-tile staging.
#define USE_ASYNC_COPY 1

typedef __attribute__((ext_vector_type(2))) float v2f;
typedef __attribute__((ext_vector_type(8))) float v8f;

// -------------------- stage 0: max over probs per rectangle --------------------
__global__ void k_maxprob(const float* __restrict__ x, float* __restrict__ mp) {
    int idx = blockIdx.x * blockDim.x + threadIdx.x;   // B*N = 2048
    if (idx >= B_ * N_) return;
    const float* p = x + (size_t)idx * (4 + CC_) + 4;
    float m = p[0];
    #pragma unroll
    for (int c = 1; c < CC_; ++c) m = fmaxf(m, p[c]);
    mp[idx] = m;
}

// -------------------- stage 1: rectangle render --------------------
__device__ __forceinline__ int clamp01(int v) { return v < 0 ? 0 : (v > 1 ? 1 : v); }

__global__ void k_render(const float* __restrict__ x, const float* __restrict__ mp,
                         float* __restrict__ layout) {
    int idx = blockIdx.x * blockDim.x + threadIdx.x;
    const int total = B_ * N_ * HH_ * WW_;             // 8,128,512
    if (idx >= total) return;
    int j = idx % WW_;
    int t = idx / WW_;
    int i = t % HH_;
    t /= HH_;
    int n = t % N_;
    int b = t / N_;
    const float* c = x + ((size_t)(b * N_ + n)) * (4 + CC_);
    int xL = (int)c[0], yT = (int)c[1], xR = (int)c[2], yB = (int)c[3];

    int t1 = clamp01(j - yT);
    int t2 = clamp01(yB - j);
    int t3 = clamp01(i - xL);
    int t4 = clamp01(xR - i);
    int e0 = 1 - abs(i - xL); if (e0 < 0) e0 = 0;
    int e1 = 1 - abs(i - xR); if (e1 < 0) e1 = 0;
    int e2 = 1 - abs(j - yT); if (e2 < 0) e2 = 0;
    int e3 = 1 - abs(j - yB); if (e3 < 0) e3 = 0;
    int F0 = e0 * t1 * t2;
    int F1 = e1 * t1 * t2;
    int F2 = e2 * t3 * t4;
    int F3 = e3 * t3 * t4;
    int val = max(max(F0, F1), max(F2, F3));           // 0 or 1
    layout[idx] = val ? mp[b * N_ + n] : 0.0f;
}

// -------------------- WMMA implicit-GEMM conv (3x3, stride 2, VALID) --------------------
// M = OC, N = B*OH*OW, K = IC*9 (must be a multiple of KC_).
// Per block: 8 waves share one M-tile; A tile staged in LDS via async global->LDS copy.
// A(16x4 f32):  lane l, reg r -> M = l&15, K = 2*(l>>4)+r
// B(4x16 f32):  lane l, reg r -> N = l&15, K = 2*(l>>4)+r
// D(16x16 f32): lane l, reg r -> N = l&15, M = r + 8*(l>>4)
__global__ void k_conv_wmma(const float* __restrict__ in, const float* __restrict__ wgt,
                            const float* __restrict__ bias, float* __restrict__ out,
                            int IC, int OC, int IH, int OH, int Ntiles) {
    __shared__ float ldsA[16 * KC_];
    const int IW = IH, OW = OH;
    const int K = IC * 9;
    const int tid = threadIdx.y * 32 + threadIdx.x;

    int ntile = blockIdx.x * blockDim.y + threadIdx.y;
    bool active = ntile < Ntiles;                      // wave-uniform

    int lane = threadIdx.x;
    int half = lane >> 4;
    int lmod = lane & 15;
    int n_idx = ntile * 16 + lmod;
    int ow = n_idx % OW;
    int tt = n_idx / OW;
    int oh = tt % OH;
    int b  = tt / OH;

    const float* wsrc = wgt + (size_t)blockIdx.y * 16 * K;
    const int kbase = half * 2;
    v8f acc = {};

    for (int k0 = 0; k0 < K; k0 += KC_) {
        // ---- stage A chunk: rows [Mbase, Mbase+16), K range [k0, k0+KC_) ----
        for (int t = tid; t < 16 * KC_; t += 256) {
            int r  = t / KC_;
            int kk = t - r * KC_;
            const float* g = wsrc + (size_t)r * K + k0 + kk;
#if USE_ASYNC_COPY
            unsigned lo = (unsigned)(uintptr_t)&ldsA[t];   // addr[31:0] = LDS offset
            asm volatile("global_load_async_to_lds_b32 %0, %1, off"
                         :: "v"(lo), "v"((unsigned long long)(uintptr_t)g)
                         : "memory");
#else
            ldsA[t] = *g;
#endif
        }
#if USE_ASYNC_COPY
        asm volatile("s_wait_asynccnt 0x0" ::: "memory");
#endif
        __syncthreads();

        if (active) {                                   // wave-uniform -> EXEC all-1s
            for (int kk = k0; kk < k0 + KC_; kk += 4) {
                int ka = kk + kbase;
                v2f A, Bv;
                A.x = ldsA[lmod * KC_ + (ka - k0)];
                A.y = ldsA[lmod * KC_ + (ka - k0) + 1];
                #pragma unroll
                for (int r = 0; r < 2; ++r) {
                    int kb  = ka + r;
                    int ic  = kb / 9;
                    int rem = kb - ic * 9;
                    int kh  = rem / 3;
                    int kw  = rem - kh * 3;
                    int ih  = oh * 2 + kh;
                    int iw  = ow * 2 + kw;
                    float v = in[(((size_t)b * IC + ic) * IH + ih) * IW + iw];
                    if (r == 0) Bv.x = v; else Bv.y = v;
                }
                acc = __builtin_amdgcn_wmma_f32_16x16x4_f32(false, A, false, Bv,
                                                            (short)0, acc, false, false);
            }
        }
        __syncthreads();
    }

    if (active) {
        #pragma unroll
        for (int r = 0; r < 8; ++r) {
            int oc = blockIdx.y * 16 + half * 8 + r;
            out[(((size_t)b * OC + oc) * OH + oh) * OW + ow] = acc[r] + bias[oc];
        }
    }
}

// -------------------- batch-norm statistics (per channel over B,H,W) --------------------
__global__ void k_bnstats(const float* __restrict__ y, float* __restrict__ mv,
                          int C, int HW) {
    int c = blockIdx.x;
    int total = B_ * HW;
    float s = 0.f, s2 = 0.f;
    for (int t = threadIdx.x; t < total; t += blockDim.x) {
        int b = t / HW;
        int p = t - b * HW;
        float v = y[((size_t)b * C + c) * HW + p];
        s += v;
        s2 += v * v;
    }
    __shared__ float sh[512];
    sh[threadIdx.x]       = s;
    sh[256 + threadIdx.x] = s2;
    __syncthreads();
    for (int st = 128; st > 0; st >>= 1) {
        if (threadIdx.x < st) {
            sh[threadIdx.x]       += sh[threadIdx.x + st];
            sh[256 + threadIdx.x] += sh[256 + threadIdx.x + st];
        }
        __syncthreads();
    }
    if (threadIdx.x == 0) {
        float inv  = 1.0f / (float)total;
        float mean = sh[0] * inv;
        float var  = sh[256] * inv - mean * mean;
        mv[c]     = mean;
        mv[C + c] = var;
    }
}

// -------------------- batch-norm apply + ReLU (in place) --------------------
__global__ void k_bnapply(float* __restrict__ y, const float* __restrict__ g,
                          const float* __restrict__ bet, const float* __restrict__ mv,
                          int C, int HW, int total) {
    int idx = blockIdx.x * blockDim.x + threadIdx.x;
    if (idx >= total) return;
    int c = (idx / HW) % C;
    float mean = mv[c], var = mv[C + c];
    float v = g[c] * (y[idx] - mean) * rsqrtf(var + 1e-5f) + bet[c];
    y[idx] = fmaxf(v, 0.0f);
}

// -------------------- FC1 via f32 WMMA (M=64 feat, N=32 batch, K=6272) + ReLU ----------
__global__ void k_fc1_wmma(const float* __restrict__ act, const float* __restrict__ wgt,
                           const float* __restrict__ bias, float* __restrict__ out,
                           int Kdim, int OF) {
    int lane = threadIdx.x;
    int half = lane >> 4;
    int lmod = lane & 15;
    int m  = blockIdx.y * 16 + lmod;   // out feature
    int nb = blockIdx.x * 16 + lmod;   // batch element
    const float* wrow = wgt + (size_t)m * Kdim;
    const float* arow = act + (size_t)nb * Kdim;
    const int kbase = half * 2;

    v8f acc = {};
    for (int kk = 0; kk < Kdim; kk += 4) {
        v2f A, Bv;
        A.x  = wrow[kk + kbase];
        A.y  = wrow[kk + kbase + 1];
        Bv.x = arow[kk + kbase];
        Bv.y = arow[kk + kbase + 1];
        acc = __builtin_amdgcn_wmma_f32_16x16x4_f32(false, A, false, Bv,
                                                    (short)0, acc, false, false);
    }
    #pragma unroll
    for (int r = 0; r < 8; ++r) {
        int o = blockIdx.y * 16 + half * 8 + r;
        float v = acc[r] + bias[o];
        out[(size_t)(blockIdx.x * 16 + lmod) * OF + o] = fmaxf(v, 0.0f);
    }
}

// -------------------- FC2 + sigmoid --------------------
__global__ void k_fc2(const float* __restrict__ h, const float* __restrict__ w,
                      const float* __restrict__ bb, float* __restrict__ out) {
    int b = threadIdx.x;
    if (b >= B_) return;
    float s = bb[0];
    #pragma unroll
    for (int o = 0; o < 64; ++o) s += h[b * 64 + o] * w[o];
    out[b] = 1.0f / (1.0f + expf(-s));
}

// ==================== host launcher ====================
extern "C" void kernel_launch(void* const* d_in, const int* in_sizes, int n_in,
                              void* d_out, int out_size, void* d_ws, size_t ws_size,
                              hipStream_t stream) {
    const float* x       = (const float*)d_in[0];
    const float* conv1_w = (const float*)d_in[1];
    const float* conv1_b = (const float*)d_in[2];
    const float* bn1_g   = (const float*)d_in[3];
    const float* bn1_b   = (const float*)d_in[4];
    const float* conv2_w = (const float*)d_in[5];
    const float* conv2_b = (const float*)d_in[6];
    const float* bn2_g   = (const float*)d_in[7];
    const float* bn2_b   = (const float*)d_in[8];
    const float* conv3_w = (const float*)d_in[9];
    const float* conv3_b = (const float*)d_in[10];
    const float* bn3_g   = (const float*)d_in[11];
    const float* bn3_b   = (const float*)d_in[12];
    const float* fc1_w   = (const float*)d_in[13];
    const float* fc1_b   = (const float*)d_in[14];
    const float* fc2_w   = (const float*)d_in[15];
    const float* fc2_b   = (const float*)d_in[16];
    float* outp = (float*)d_out;

    char* ws = (char*)d_ws;
    size_t off = 0;
    auto take = [&](size_t bytes) -> void* {
        off = (off + 255) & ~(size_t)255;
        void* p = ws + off;
        off += bytes;
        return p;
    };
    float* layout = (float*)take((size_t)B_ * N_ * HH_ * WW_ * 4);        // (32,64,63,63)
    float* mp     = (float*)take((size_t)B_ * N_ * 4);
    float* a1     = (float*)take((size_t)B_ * 64  * 31 * 31 * 4);         // conv1 out
    float* a2     = (float*)take((size_t)B_ * 128 * 15 * 15 * 4);         // conv2 out
    float* a3     = (float*)take((size_t)B_ * 128 * 7  * 7  * 4);         // conv3 out
    float* mv     = (float*)take(2 * 128 * 4);                            // mean|var scratch
    float* h1     = (float*)take((size_t)B_ * 64 * 4);                    // fc1 out

    // stage 0/1: render
    k_maxprob<<<(B_ * N_ + 255) / 256, 256, 0, stream>>>(x, mp);
    {
        int total = B_ * N_ * HH_ * WW_;
        k_render<<<(total + 255) / 256, 256, 0, stream>>>(x, mp, layout);
    }

    // conv1: IC=64 OC=64 63->31
    {
        int OH = 31, Ntiles = B_ * OH * OH / 16;                          // 1922
        dim3 grid((Ntiles + 7) / 8, 64 / 16), blk(32, 8);
        k_conv_wmma<<<grid, blk, 0, stream>>>(layout, conv1_w, conv1_b, a1, 64, 64, 63, OH, Ntiles);
        int HW = OH * OH, total = B_ * 64 * HW;
        k_bnstats<<<64, 256, 0, stream>>>(a1, mv, 64, HW);
        k_bnapply<<<(total + 255) / 256, 256, 0, stream>>>(a1, bn1_g, bn1_b, mv, 64, HW, total);
    }
    // conv2: IC=64 OC=128 31->15
    {
        int OH = 15, Ntiles = B_ * OH * OH / 16;                          // 450
        dim3 grid((Ntiles + 7) / 8, 128 / 16), blk(32, 8);
        k_conv_wmma<<<grid, blk, 0, stream>>>(a1, conv2_w, conv2_b, a2, 64, 128, 31, OH, Ntiles);
        int HW = OH * OH, total = B_ * 128 * HW;
        k_bnstats<<<128, 256, 0, stream>>>(a2, mv, 128, HW);
        k_bnapply<<<(total + 255) / 256, 256, 0, stream>>>(a2, bn2_g, bn2_b, mv, 128, HW, total);
    }
    // conv3: IC=128 OC=128 15->7  (K=1152 -> two KC_ chunks)
    {
        int OH = 7, Ntiles = B_ * OH * OH / 16;                           // 98
        dim3 grid((Ntiles + 7) / 8, 128 / 16), blk(32, 8);
        k_conv_wmma<<<grid, blk, 0, stream>>>(a2, conv3_w, conv3_b, a3, 128, 128, 15, OH, Ntiles);
        int HW = OH * OH, total = B_ * 128 * HW;
        k_bnstats<<<128, 256, 0, stream>>>(a3, mv, 128, HW);
        k_bnapply<<<(total + 255) / 256, 256, 0, stream>>>(a3, bn3_g, bn3_b, mv, 128, HW, total);
    }
    // fc1: (32,6272) @ (64,6272)^T -> (32,64), ReLU
    {
        dim3 grid(B_ / 16, 64 / 16), blk(32, 1);
        k_fc1_wmma<<<grid, blk, 0, stream>>>(a3, fc1_w, fc1_b, h1, 128 * 49, 64);
    }
    // fc2 + sigmoid -> (32,1)
    k_fc2<<<1, 32, 0, stream>>>(h1, fc2_w, fc2_b, outp);

    (void)in_sizes; (void)n_in; (void)out_size; (void)ws_size;
}